// KGCN_kg_37950331028018
// MI455X (gfx1250) — compile-verified
//
#include <hip/hip_runtime.h>

#define D    128
#define K    16
#define PAD  132          // LDS row stride in floats: (row*132 + col) % 64 banks -> conflict-free
#define BSZ  2048

typedef __attribute__((ext_vector_type(2))) float v2f;
typedef __attribute__((ext_vector_type(4))) float v4f;
typedef __attribute__((ext_vector_type(8))) float v8f;

__device__ __forceinline__ float sigmoidf_(float x) { return 1.0f / (1.0f + __expf(-x)); }

// Async-copy W (global row-major [128][128]) into padded LDS [128][PAD] via
// GLOBAL_LOAD_ASYNC_TO_LDS_B128 (ASYNCcnt). No VGPR round-trip; completion is
// deferred until wait_async_lds() right before the WMMA phase, so the whole
// attention/softmax/gather phase overlaps the 64KB L2->LDS fetch.
__device__ __forceinline__ void load_W_async(const float* __restrict__ Wg, float* Wlds) {
#pragma unroll
    for (int it = 0; it < 16; ++it) {
        const int i = threadIdx.x + it * 256;       // 4096 16B chunks total
        const int r = i >> 5;                        // 32 float4 per row
        const int c = (i & 31) * 4;
        const unsigned lds_off = (unsigned)(unsigned long long)(const void*)(Wlds + r * PAD + c);
        const unsigned gl_off  = (unsigned)((r * D + c) * sizeof(float));
        asm volatile("global_load_async_to_lds_b128 %0, %1, %2"
                     :
                     : "v"(lds_off), "v"(gl_off), "s"(Wg)
                     : "memory");
    }
}

__device__ __forceinline__ void wait_async_lds() {
    asm volatile("s_wait_asynccnt 0x0" ::: "memory");
}

// One wave computes a 16x16 output tile: OUT[m][j0+n] = sum_d X[m][d] * W[j0+n][d]
// via 32 chained V_WMMA_F32_16X16X4_F32 (K=128 in steps of 4), fp32 exact.
// A frag (16x4 f32): lane l holds A[l&15][2*(l>>4) + i] in a[i]
// B frag (4x16 f32): lane l holds B[2*(l>>4) + i][l&15]  = W[j0+(l&15)][k + ...]
// D frag (16x16 f32): lane l, vgpr v holds OUT[v + 8*(l>>4)][l&15]
__device__ __forceinline__ v8f linear_tile(const float* Wlds, const float* xlds,
                                           int lane, int j0) {
    const int rowA = lane & 15;
    const int koff = (lane >> 4) * 2;
    v8f acc = {};
#pragma unroll
    for (int kc = 0; kc < D / 4; ++kc) {
        const int k0 = kc * 4 + koff;
        v2f a = *(const v2f*)(xlds + rowA * PAD + k0);
        v2f b = *(const v2f*)(Wlds + (j0 + rowA) * PAD + k0);
        acc = __builtin_amdgcn_wmma_f32_16x16x4_f32(false, a, false, b,
                                                    (short)0, acc, false, false);
    }
    return acc;
}

// -------- iter0 / hop1: one block per batch element b; 16 rows = hop-1 entities --------
__global__ __launch_bounds__(256, 1) void kgcn_hop1(
    const float* __restrict__ usr_w, const float* __restrict__ ent_w,
    const float* __restrict__ rel_w, const float* __restrict__ Wg,
    const float* __restrict__ bias, const int* __restrict__ user_ids,
    const int* __restrict__ item_ids, const int* __restrict__ adj_ent,
    const int* __restrict__ adj_rel, float* __restrict__ ev1) {
    __shared__ float Wlds[D * PAD];
    __shared__ float xlds[16 * PAD];
    __shared__ float ulds[D];
    __shared__ float attnS[16][K];
    __shared__ int   e1S[16];
    __shared__ int   e2S[16][K];

    const int b   = blockIdx.x;
    const int tid = threadIdx.x;

    load_W_async(Wg, Wlds);                      // overlap with attention phase
    if (tid < D)  ulds[tid] = usr_w[user_ids[b] * D + tid];
    if (tid < 16) e1S[tid]  = adj_ent[item_ids[b] * K + tid];
    __syncthreads();

    // attention scores: thread (m,k) = (tid>>4, tid&15)
    const int m  = tid >> 4;
    const int kk = tid & 15;
    const int e1 = e1S[m];
    const int rix = adj_rel[e1 * K + kk];
    e2S[m][kk] = adj_ent[e1 * K + kk];
    float s = 0.f;
    const float* rr = rel_w + rix * D;
#pragma unroll 8
    for (int d = 0; d < D; ++d) s += ulds[d] * rr[d];
    attnS[m][kk] = s;
    __syncthreads();
    float mx = attnS[m][0];
#pragma unroll
    for (int k2 = 1; k2 < K; ++k2) mx = fmaxf(mx, attnS[m][k2]);
    float sum = 0.f;
#pragma unroll
    for (int k2 = 0; k2 < K; ++k2) sum += __expf(attnS[m][k2] - mx);
    const float myp = __expf(s - mx) / sum;
    __syncthreads();
    attnS[m][kk] = myp;
    __syncthreads();

    // aggregation: wave w handles rows 2w, 2w+1
    const int wave = tid >> 5, lane = tid & 31;
    for (int rm = wave * 2; rm < wave * 2 + 2; ++rm) {
        float at[K]; int ix[K];
#pragma unroll
        for (int k2 = 0; k2 < K; ++k2) { at[k2] = attnS[rm][k2]; ix[k2] = e2S[rm][k2]; }
        const int e1r = e1S[rm];
        for (int dd = lane; dd < D; dd += 32) {
            float acc = ent_w[e1r * D + dd];
#pragma unroll
            for (int k2 = 0; k2 < K; ++k2) acc += at[k2] * ent_w[ix[k2] * D + dd];
            xlds[rm * PAD + dd] = acc;
        }
    }
    wait_async_lds();                            // W resident in LDS
    __syncthreads();

    // linear + sigmoid
    const int j0 = wave * 16;
    v8f acc = linear_tile(Wlds, xlds, lane, j0);
    const int jc = j0 + (lane & 15);
    const float bj = bias[jc];
#pragma unroll
    for (int v = 0; v < 8; ++v) {
        const int rm = v + ((lane >> 4) << 3);
        ev1[(b * 16 + rm) * D + jc] = sigmoidf_(acc[v] + bj);
    }
}

// -------- iter0 / hop0: one block per 16 batch elements; row m <-> b' = 16*blk+m --------
__global__ __launch_bounds__(256, 1) void kgcn_hop0_it0(
    const float* __restrict__ usr_w, const float* __restrict__ ent_w,
    const float* __restrict__ rel_w, const float* __restrict__ Wg,
    const float* __restrict__ bias, const int* __restrict__ user_ids,
    const int* __restrict__ item_ids, const int* __restrict__ adj_ent,
    const int* __restrict__ adj_rel, float* __restrict__ ev0) {
    __shared__ float Wlds[D * PAD];
    __shared__ float xlds[16 * PAD];
    __shared__ float ulds[16][PAD];
    __shared__ float attnS[16][K];
    __shared__ int   itemS[16];
    __shared__ int   e2S[16][K];

    const int rowbase = blockIdx.x * 16;
    const int tid = threadIdx.x;

    load_W_async(Wg, Wlds);
    for (int i = tid; i < 16 * D; i += 256) {
        const int r = i >> 7, c = i & 127;
        ulds[r][c] = usr_w[user_ids[rowbase + r] * D + c];
    }
    if (tid < 16) itemS[tid] = item_ids[rowbase + tid];
    __syncthreads();

    const int m  = tid >> 4;
    const int kk = tid & 15;
    const int it = itemS[m];
    const int rix = adj_rel[it * K + kk];
    e2S[m][kk] = adj_ent[it * K + kk];
    float s = 0.f;
    const float* rr = rel_w + rix * D;
#pragma unroll 8
    for (int d = 0; d < D; ++d) s += ulds[m][d] * rr[d];
    attnS[m][kk] = s;
    __syncthreads();
    float mx = attnS[m][0];
#pragma unroll
    for (int k2 = 1; k2 < K; ++k2) mx = fmaxf(mx, attnS[m][k2]);
    float sum = 0.f;
#pragma unroll
    for (int k2 = 0; k2 < K; ++k2) sum += __expf(attnS[m][k2] - mx);
    const float myp = __expf(s - mx) / sum;
    __syncthreads();
    attnS[m][kk] = myp;
    __syncthreads();

    const int wave = tid >> 5, lane = tid & 31;
    for (int rm = wave * 2; rm < wave * 2 + 2; ++rm) {
        float at[K]; int ix[K];
#pragma unroll
        for (int k2 = 0; k2 < K; ++k2) { at[k2] = attnS[rm][k2]; ix[k2] = e2S[rm][k2]; }
        const int e1r = itemS[rm];
        for (int dd = lane; dd < D; dd += 32) {
            float acc = ent_w[e1r * D + dd];
#pragma unroll
            for (int k2 = 0; k2 < K; ++k2) acc += at[k2] * ent_w[ix[k2] * D + dd];
            xlds[rm * PAD + dd] = acc;
        }
    }
    wait_async_lds();
    __syncthreads();

    const int j0 = wave * 16;
    v8f acc = linear_tile(Wlds, xlds, lane, j0);
    const int jc = j0 + (lane & 15);
    const float bj = bias[jc];
#pragma unroll
    for (int v = 0; v < 8; ++v) {
        const int rm = v + ((lane >> 4) << 3);
        ev0[(rowbase + rm) * D + jc] = sigmoidf_(acc[v] + bj);
    }
}

// -------- iter1 / hop0 + final dot: one block per 16 batch elements --------
__global__ __launch_bounds__(256, 1) void kgcn_it1_final(
    const float* __restrict__ usr_w, const float* __restrict__ rel_w,
    const float* __restrict__ Wg, const float* __restrict__ bias,
    const int* __restrict__ user_ids, const int* __restrict__ item_ids,
    const int* __restrict__ adj_rel, const float* __restrict__ ev0,
    const float* __restrict__ ev1, float* __restrict__ out) {
    __shared__ float Wlds[D * PAD];
    __shared__ float xlds[16 * PAD];
    __shared__ float ulds[16][PAD];
    __shared__ float olds[16][PAD];
    __shared__ float attnS[16][K];
    __shared__ int   itemS[16];

    const int rowbase = blockIdx.x * 16;
    const int tid = threadIdx.x;

    load_W_async(Wg, Wlds);
    for (int i = tid; i < 16 * D; i += 256) {
        const int r = i >> 7, c = i & 127;
        ulds[r][c] = usr_w[user_ids[rowbase + r] * D + c];
    }
    if (tid < 16) itemS[tid] = item_ids[rowbase + tid];
    __syncthreads();

    const int m  = tid >> 4;
    const int kk = tid & 15;
    const int rix = adj_rel[itemS[m] * K + kk];   // rv[0] reused in iteration 1
    float s = 0.f;
    const float* rr = rel_w + rix * D;
#pragma unroll 8
    for (int d = 0; d < D; ++d) s += ulds[m][d] * rr[d];
    attnS[m][kk] = s;
    __syncthreads();
    float mx = attnS[m][0];
#pragma unroll
    for (int k2 = 1; k2 < K; ++k2) mx = fmaxf(mx, attnS[m][k2]);
    float sum = 0.f;
#pragma unroll
    for (int k2 = 0; k2 < K; ++k2) sum += __expf(attnS[m][k2] - mx);
    const float myp = __expf(s - mx) / sum;
    __syncthreads();
    attnS[m][kk] = myp;
    __syncthreads();

    const int wave = tid >> 5, lane = tid & 31;
    for (int rm = wave * 2; rm < wave * 2 + 2; ++rm) {
        float at[K];
#pragma unroll
        for (int k2 = 0; k2 < K; ++k2) at[k2] = attnS[rm][k2];
        const int bp = rowbase + rm;
        for (int dd = lane; dd < D; dd += 32) {
            float acc = ev0[bp * D + dd];
#pragma unroll
            for (int k2 = 0; k2 < K; ++k2) acc += at[k2] * ev1[(bp * 16 + k2) * D + dd];
            xlds[rm * PAD + dd] = acc;
        }
    }
    wait_async_lds();
    __syncthreads();

    const int j0 = wave * 16;
    v8f acc = linear_tile(Wlds, xlds, lane, j0);
    const int jc = j0 + (lane & 15);
    const float bj = bias[jc];
#pragma unroll
    for (int v = 0; v < 8; ++v) {
        const int rm = v + ((lane >> 4) << 3);
        olds[rm][jc] = tanhf(acc[v] + bj);
    }
    __syncthreads();

    // final: out[b'] = sigmoid(user . item)
    if (tid < 16) {
        float dot = 0.f;
#pragma unroll 8
        for (int d = 0; d < D; ++d) dot += ulds[tid][d] * olds[tid][d];
        out[rowbase + tid] = sigmoidf_(dot);
    }
}

extern "C" void kernel_launch(void* const* d_in, const int* in_sizes, int n_in,
                              void* d_out, int out_size, void* d_ws, size_t ws_size,
                              hipStream_t stream) {
    const float* usr_w    = (const float*)d_in[0];
    const float* ent_w    = (const float*)d_in[1];
    const float* rel_w    = (const float*)d_in[2];
    const float* Wg       = (const float*)d_in[3];
    const float* bias     = (const float*)d_in[4];
    const int*   user_ids = (const int*)d_in[5];
    const int*   item_ids = (const int*)d_in[6];
    const int*   adj_ent  = (const int*)d_in[7];
    const int*   adj_rel  = (const int*)d_in[8];
    float* out = (float*)d_out;

    float* ev1 = (float*)d_ws;                 // [B*16*D]
    float* ev0 = ev1 + (size_t)BSZ * 16 * D;   // [B*D]

    kgcn_hop1<<<BSZ, 256, 0, stream>>>(usr_w, ent_w, rel_w, Wg, bias,
                                       user_ids, item_ids, adj_ent, adj_rel, ev1);
    kgcn_hop0_it0<<<BSZ / 16, 256, 0, stream>>>(usr_w, ent_w, rel_w, Wg, bias,
                                                user_ids, item_ids, adj_ent, adj_rel, ev0);
    kgcn_it1_final<<<BSZ / 16, 256, 0, stream>>>(usr_w, rel_w, Wg, bias,
                                                 user_ids, item_ids, adj_rel,
                                                 ev0, ev1, out);
}